// MultiBoxHeads_83734682403238
// MI455X (gfx1250) — compile-verified
//
#include <hip/hip_runtime.h>
#include <stdint.h>

#define B_ 8
#define P_ 131072
#define C_ 21
#define CM1 20
#define PRE 1024
#define TOPK 100
#define CAP 262144
#define SCORE_TH 0.1f
#define NMS_TH 0.45f

typedef float v2f __attribute__((ext_vector_type(2)));
typedef float v8f __attribute__((ext_vector_type(8)));

// ---------------------------------------------------------------------------
// Stage 0: zero per-batch candidate counters
// ---------------------------------------------------------------------------
__global__ void zero_counts(int* cnt) {
    if (threadIdx.x < B_) cnt[threadIdx.x] = 0;
}

// ---------------------------------------------------------------------------
// Stage 1: softmax + threshold + compact candidates.
// conf rows are 21 floats -> stage a 256-prior tile (21504 B, 16B-aligned since
// 256*21*4 % 16 == 0) into LDS with gfx1250 async-to-LDS b128 ops, then each
// thread does a register softmax from LDS (stride 21 is odd -> conflict-free).
// ---------------------------------------------------------------------------
__global__ __launch_bounds__(256) void score_kernel(
    const float* __restrict__ conf,
    float* __restrict__ cscore, uint32_t* __restrict__ cidx,
    int* __restrict__ ccnt)
{
    __shared__ float sconf[256 * C_];      // 21504 bytes
    const int b  = blockIdx.y;
    const int p0 = blockIdx.x * 256;
    const int t  = threadIdx.x;
    const float* gsrc = conf + ((size_t)b * P_ + p0) * C_;

    // 256*21 floats = 1344 x 16-byte async copies into LDS
    #pragma unroll
    for (int k = 0; k < 6; ++k) {
        int i = t + k * 256;
        if (i < (256 * C_) / 4) {
            unsigned lds = (unsigned)(uintptr_t)(&sconf[i * 4]);
            unsigned long long ga = (unsigned long long)(uintptr_t)(gsrc + i * 4);
            asm volatile("global_load_async_to_lds_b128 %0, %1, off"
                         :: "v"(lds), "v"(ga) : "memory");
        }
    }
    asm volatile("s_wait_asynccnt 0" ::: "memory");
    __syncthreads();

    const float* my = &sconf[t * C_];
    float m = my[0];
    #pragma unroll
    for (int c = 1; c < C_; ++c) m = fmaxf(m, my[c]);
    float e[C_];
    float s = 0.f;
    #pragma unroll
    for (int c = 0; c < C_; ++c) { e[c] = __expf(my[c] - m); s += e[c]; }
    const float inv = 1.0f / s;

    const int p = p0 + t;
    #pragma unroll
    for (int c = 1; c < C_; ++c) {
        float pr = e[c] * inv;
        if (pr > SCORE_TH) {
            int slot = atomicAdd(&ccnt[b], 1);
            if (slot < CAP) {
                cscore[(size_t)b * CAP + slot] = pr;
                cidx[(size_t)b * CAP + slot]   = (uint32_t)(p * CM1 + (c - 1));
            }
        }
    }
}

// ---------------------------------------------------------------------------
// Stage 2: per-batch top-1024 (histogram select on float bits, scores are in
// (0.1, 1.0] so raw bits are monotone), bitonic sort descending (stable via
// index tie-break), box decode, class-offset boxes.
// ---------------------------------------------------------------------------
__global__ __launch_bounds__(1024) void topk_kernel(
    const float* __restrict__ loc, const float* __restrict__ priors,
    const float* __restrict__ cscore, const uint32_t* __restrict__ cidx,
    const int* __restrict__ ccnt,
    float* __restrict__ tscore, uint32_t* __restrict__ tidx,
    float* __restrict__ cboxes, float* __restrict__ oboxes)
{
    __shared__ unsigned hist[1024];
    __shared__ float    ss[1024];
    __shared__ uint32_t si[1024];
    __shared__ float    red[1024];
    __shared__ int s_tb, s_hi, s_c0, s_c1;

    const int b = blockIdx.x;
    const int t = threadIdx.x;
    int n = ccnt[b]; if (n > CAP) n = CAP;
    const int K = (n < PRE) ? n : PRE;

    hist[t] = 0;
    if (t == 0) { s_c0 = 0; s_c1 = 0; }
    __syncthreads();

    const unsigned LO = 0x3DCCCCCCu, HI = 0x3F800000u;  // bits(0.1) .. bits(1.0)
    const float*    cs = cscore + (size_t)b * CAP;
    const uint32_t* ci = cidx   + (size_t)b * CAP;

    for (int i = t; i < n; i += 1024) {
        unsigned bits = __float_as_uint(cs[i]);
        unsigned bk = (unsigned)(((unsigned long long)(bits - LO) * 1024ull)
                                 / (unsigned long long)(HI - LO + 1));
        if (bk > 1023u) bk = 1023u;
        atomicAdd(&hist[bk], 1u);
    }
    __syncthreads();

    // inclusive suffix scan (Hillis-Steele)
    for (int off = 1; off < 1024; off <<= 1) {
        unsigned v = hist[t];
        unsigned a = (t + off < 1024) ? hist[t + off] : 0u;
        __syncthreads();
        hist[t] = v + a;
        __syncthreads();
    }

    if (K > 0) {
        unsigned sfx = hist[t];
        unsigned nxt = (t < 1023) ? hist[t + 1] : 0u;
        if (sfx >= (unsigned)K && nxt < (unsigned)K) { s_tb = t; s_hi = (int)nxt; }
    } else if (t == 0) { s_tb = 1024; s_hi = 0; }
    ss[t] = 0.0f; si[t] = 0u;
    __syncthreads();

    const int tb = s_tb, hi = s_hi;
    for (int i = t; i < n; i += 1024) {
        float sc = cs[i];
        unsigned bits = __float_as_uint(sc);
        unsigned bk = (unsigned)(((unsigned long long)(bits - LO) * 1024ull)
                                 / (unsigned long long)(HI - LO + 1));
        if (bk > 1023u) bk = 1023u;
        int dst = -1;
        if ((int)bk > tb)       dst = atomicAdd(&s_c0, 1);
        else if ((int)bk == tb) dst = hi + atomicAdd(&s_c1, 1);
        if (dst >= 0 && dst < K) { ss[dst] = sc; si[dst] = ci[i]; }
    }
    __syncthreads();

    // bitonic sort, descending by score, ascending by idx on ties
    for (int k2 = 2; k2 <= 1024; k2 <<= 1) {
        for (int j = k2 >> 1; j > 0; j >>= 1) {
            int ixj = t ^ j;
            if (ixj > t) {
                float a = ss[t], c2 = ss[ixj];
                uint32_t ia = si[t], ic = si[ixj];
                bool desc = ((t & k2) == 0);
                bool a_first = (a > c2) || (a == c2 && ia <= ic);
                bool doswap = desc ? (!a_first) : a_first;
                if (doswap) { ss[t] = c2; ss[ixj] = a; si[t] = ic; si[ixj] = ia; }
            }
            __syncthreads();
        }
    }

    // decode boxes for the selected candidates
    float sc = ss[t];
    uint32_t id = si[t];
    float bx[4] = {0.f, 0.f, 0.f, 0.f};
    int lab = 0;
    if (sc > 0.0f) {
        int prior = (int)(id / CM1);
        lab = (int)(id % CM1) + 1;
        const float* lp = loc    + ((size_t)b * P_ + prior) * 4;
        const float* pp = priors + (size_t)prior * 4;
        float cx = pp[0] + lp[0] * 0.1f * pp[2];
        float cy = pp[1] + lp[1] * 0.1f * pp[3];
        float w  = pp[2] * __expf(lp[2] * 0.2f);
        float h  = pp[3] * __expf(lp[3] * 0.2f);
        bx[0] = cx - 0.5f * w; bx[1] = cy - 0.5f * h;
        bx[2] = cx + 0.5f * w; bx[3] = cy + 0.5f * h;
    }
    tscore[(size_t)b * PRE + t] = sc;
    tidx[(size_t)b * PRE + t]   = id;
    #pragma unroll
    for (int c = 0; c < 4; ++c)
        cboxes[((size_t)b * PRE + t) * 4 + c] = bx[c];

    // max over all candidate box coords (for per-class offsetting)
    red[t] = fmaxf(fmaxf(bx[0], bx[1]), fmaxf(bx[2], bx[3]));
    __syncthreads();
    for (int off = 512; off > 0; off >>= 1) {
        if (t < off) red[t] = fmaxf(red[t], red[t + off]);
        __syncthreads();
    }
    float shift = (float)lab * (red[0] + 1.0f);
    #pragma unroll
    for (int c = 0; c < 4; ++c)
        oboxes[((size_t)b * PRE + t) * 4 + c] = bx[c] + shift;
}

// ---------------------------------------------------------------------------
// Stage 3: 1024x1024 pairwise IoU per batch. One wave per 16x16 tile.
// The rank-2 term (area_i + area_j) is computed on the matrix unit with
// V_WMMA_F32_16X16X4_F32: A=[area_row,1] (16x4), B=[1;area_col] (4x16).
// VALU does only the min/max intersection. D layout: lane 0-15 -> M=v,
// lane 16-31 -> M=v+8, N = lane&15 (8 VGPRs per lane).
// ---------------------------------------------------------------------------
__global__ __launch_bounds__(256) void iou_kernel(
    const float* __restrict__ oboxes, float* __restrict__ iou)
{
    __shared__ float sb[8][32][4];
    const int b    = blockIdx.y;
    const int w    = threadIdx.x >> 5;
    const int lane = threadIdx.x & 31;
    const int tile = blockIdx.x * 8 + w;
    const int tm = tile >> 6, tn = tile & 63;

    // lanes 0-15 load the 16 row boxes, lanes 16-31 the 16 column boxes
    int r = (lane < 16) ? (tm * 16 + lane) : (tn * 16 + (lane - 16));
    const float* src = oboxes + ((size_t)b * PRE + r) * 4;
    sb[w][lane][0] = src[0]; sb[w][lane][1] = src[1];
    sb[w][lane][2] = src[2]; sb[w][lane][3] = src[3];
    __syncthreads();

    float arow = 0.f, acol = 0.f;
    if (lane < 16) {
        arow = (sb[w][lane][2]      - sb[w][lane][0])      * (sb[w][lane][3]      - sb[w][lane][1]);
        acol = (sb[w][16 + lane][2] - sb[w][16 + lane][0]) * (sb[w][16 + lane][3] - sb[w][16 + lane][1]);
    }
    // A 16x4 f32: lanes 0-15 hold K=0,1; lanes 16-31 hold K=2,3 (zero-padded)
    v2f a, bm;
    a.x  = (lane < 16) ? arow : 0.f;  a.y  = (lane < 16) ? 1.f  : 0.f;
    bm.x = (lane < 16) ? 1.f  : 0.f;  bm.y = (lane < 16) ? acol : 0.f;
    v8f c = {};
    v8f d = __builtin_amdgcn_wmma_f32_16x16x4_f32(
        /*neg_a=*/false, a, /*neg_b=*/false, bm,
        /*c_mod=*/(short)0, c, /*reuse_a=*/false, /*reuse_b=*/false);

    const int N = lane & 15;
    const int mbase = (lane < 16) ? 0 : 8;
    #pragma unroll
    for (int v = 0; v < 8; ++v) {
        int M = mbase + v;
        float x0 = fmaxf(sb[w][M][0], sb[w][16 + N][0]);
        float y0 = fmaxf(sb[w][M][1], sb[w][16 + N][1]);
        float x1 = fminf(sb[w][M][2], sb[w][16 + N][2]);
        float y1 = fminf(sb[w][M][3], sb[w][16 + N][3]);
        float inter = fmaxf(x1 - x0, 0.f) * fmaxf(y1 - y0, 0.f);
        float denom = d[v] - inter + 1e-12f;   // area_i + area_j - inter
        iou[((size_t)b * PRE + (tm * 16 + M)) * PRE + (tn * 16 + N)] = inter / denom;
    }
}

// ---------------------------------------------------------------------------
// Stage 4: greedy NMS (serial over i, parallel suppression over j), scan,
// scatter top-100 with clamp/validity masking.
// ---------------------------------------------------------------------------
__global__ __launch_bounds__(1024) void nms_kernel(
    const float* __restrict__ tscore, const uint32_t* __restrict__ tidx,
    const float* __restrict__ cboxes, const float* __restrict__ iou,
    float* __restrict__ out)
{
    __shared__ int keep[1024];
    __shared__ int scan[1024];
    const int b = blockIdx.x;
    const int t = threadIdx.x;

    float sc = tscore[(size_t)b * PRE + t];
    keep[t] = (sc > 0.f) ? 1 : 0;
    __syncthreads();

    const float* irow = iou + (size_t)b * PRE * PRE;
    for (int i = 0; i < PRE; ++i) {
        if (keep[i]) {
            if (t > i && irow[(size_t)i * PRE + t] > NMS_TH) keep[t] = 0;
        }
        __syncthreads();
    }

    int k = keep[t];
    scan[t] = k;
    __syncthreads();
    for (int off = 1; off < 1024; off <<= 1) {
        int v = scan[t];
        int a = (t >= off) ? scan[t - off] : 0;
        __syncthreads();
        scan[t] = v + a;
        __syncthreads();
    }
    int kcnt = scan[1023];
    int rank = scan[t] - 1;

    float* outB = out;
    float* outS = out + (size_t)B_ * TOPK * 4;
    int*   outL = (int*)(out + (size_t)B_ * TOPK * 4 + (size_t)B_ * TOPK);

    if (k && rank < TOPK) {
        uint32_t id = tidx[(size_t)b * PRE + t];
        int lab = (int)(id % CM1) + 1;
        bool valid = (sc >= SCORE_TH);
        float vmul = valid ? 1.f : 0.f;
        #pragma unroll
        for (int c = 0; c < 4; ++c) {
            float bx = cboxes[((size_t)b * PRE + t) * 4 + c];
            bx = fminf(fmaxf(bx, 0.f), 1.f);
            outB[((size_t)b * TOPK + rank) * 4 + c] = bx * vmul;
        }
        outS[(size_t)b * TOPK + rank] = sc * vmul;
        outL[(size_t)b * TOPK + rank] = valid ? lab : 0;
    }
    int kc = (kcnt > TOPK) ? TOPK : kcnt;
    if (t >= kc && t < TOPK) {
        #pragma unroll
        for (int c = 0; c < 4; ++c)
            outB[((size_t)b * TOPK + t) * 4 + c] = 0.f;
        outS[(size_t)b * TOPK + t] = 0.f;
        outL[(size_t)b * TOPK + t] = 0;
    }
}

// ---------------------------------------------------------------------------
extern "C" void kernel_launch(void* const* d_in, const int* in_sizes, int n_in,
                              void* d_out, int out_size, void* d_ws, size_t ws_size,
                              hipStream_t stream)
{
    const float* loc    = (const float*)d_in[0];
    const float* conf   = (const float*)d_in[1];
    const float* priors = (const float*)d_in[2];

    char* ws = (char*)d_ws;
    size_t off = 0;
    int*      ccnt   = (int*)(ws + off);      off += 256;
    float*    cscore = (float*)(ws + off);    off += (size_t)B_ * CAP * 4;
    uint32_t* cidx   = (uint32_t*)(ws + off); off += (size_t)B_ * CAP * 4;
    float*    tscore = (float*)(ws + off);    off += (size_t)B_ * PRE * 4;
    uint32_t* tidx   = (uint32_t*)(ws + off); off += (size_t)B_ * PRE * 4;
    float*    cboxes = (float*)(ws + off);    off += (size_t)B_ * PRE * 16;
    float*    oboxes = (float*)(ws + off);    off += (size_t)B_ * PRE * 16;
    float*    ioum   = (float*)(ws + off);    off += (size_t)B_ * PRE * PRE * 4;

    zero_counts<<<1, 32, 0, stream>>>(ccnt);

    dim3 g1(P_ / 256, B_);
    score_kernel<<<g1, 256, 0, stream>>>(conf, cscore, cidx, ccnt);

    topk_kernel<<<B_, 1024, 0, stream>>>(loc, priors, cscore, cidx, ccnt,
                                         tscore, tidx, cboxes, oboxes);

    dim3 g3((PRE / 16) * (PRE / 16) / 8, B_);   // 512 x 8
    iou_kernel<<<g3, 256, 0, stream>>>(oboxes, ioum);

    nms_kernel<<<B_, 1024, 0, stream>>>(tscore, tidx, cboxes, ioum, (float*)d_out);
}